// GCmapCrys_13640816132255
// MI455X (gfx1250) — compile-verified
//
#include <hip/hip_runtime.h>
#include <hip/hip_bf16.h>
#include <math.h>

// ---------------------------------------------------------------------------
// GNN (edge-update + GAT) forward for MI455X / gfx1250, wave32.
// Big GEMMs run on the fp32 WMMA pipe (v_wmma_f32_16x16x4_f32) to keep exact
// f32 semantics vs the JAX reference. Everything else is bandwidth-bound
// gather/scatter tuned for coalescing; scatters use f32 atomics that land in
// the 192MB L2 (accumulators are ~41MB).
// ---------------------------------------------------------------------------

#define N_NODES 20000
#define N_EDGES 160000
#define NGRAPH  64
#define HDIM    512     // H*C = 4*128
#define OE      32      // edge-feature width after layer 0
#define BN_EPS  1e-5f

typedef float v2f __attribute__((ext_vector_type(2)));
typedef float v8f __attribute__((ext_vector_type(8)));

static __device__ __forceinline__ void atomicMaxF(float* addr, float val) {
  int* ia = (int*)addr;
  int old = __float_as_int(*addr);
  while (__int_as_float(old) < val) {
    int assumed = old;
    old = atomicCAS(ia, assumed, __float_as_int(val));
    if (old == assumed) break;
  }
}

// ------------------------------ generic fill -------------------------------
__global__ void fill_k(float* __restrict__ p, float v, long long n) {
  long long i = (long long)blockIdx.x * 256 + threadIdx.x;
  if (i < n) p[i] = v;
}

// ------------------------- h0 = [emb[x_emb] | x | 0pad] --------------------
__global__ void build_h0_k(const float* __restrict__ x, const int* __restrict__ xe,
                           const float* __restrict__ emb, float* __restrict__ h0) {
  long long i = (long long)blockIdx.x * 256 + threadIdx.x;
  if (i >= (long long)N_NODES * 640) return;
  int n = (int)(i / 640), c = (int)(i % 640);
  float v;
  if (c < 32)        v = emb[xe[n] * 32 + c];
  else if (c < 622)  v = x[(long long)n * 590 + (c - 32)];
  else               v = 0.f;
  h0[i] = v;
}

// ---------------- pack [Wd|Ws] (euw rows) into KPAD x 64, zero-padded ------
__global__ void pack_w2_k(const float* __restrict__ euw, int in_e, int in_n,
                          int kpad, float* __restrict__ w2) {
  int i = blockIdx.x * 256 + threadIdx.x;
  if (i >= kpad * 64) return;
  int k = i >> 6, c = i & 63;
  float v = 0.f;
  if (k < in_n)
    v = (c < 32) ? euw[(in_e + k) * 32 + c]
                 : euw[(in_e + in_n + k) * 32 + (c - 32)];
  w2[i] = v;
}

// ---------------- pack gw into KPAD x 512, zero-padded ---------------------
__global__ void pack_gw_k(const float* __restrict__ gw, int in_n, int kpad,
                          float* __restrict__ out) {
  long long i = (long long)blockIdx.x * 256 + threadIdx.x;
  if (i >= (long long)kpad * HDIM) return;
  int k = (int)(i >> 9);
  int n = (int)(i & 511);
  out[i] = (k < in_n) ? gw[(long long)k * HDIM + n] : 0.f;
}

// ---------------------- fp32 WMMA GEMM: C[M,N] = A[M,K] @ B[K,N] -----------
// K must be a multiple of 32. 128 threads = 4 waves; 64x64 C tile per block.
__global__ __launch_bounds__(128) void gemm_f32_wmma(
    const float* __restrict__ A, int lda,
    const float* __restrict__ B, int ldb,
    float* __restrict__ C, int ldc,
    int M, int Ncols, int K) {
  __shared__ float As[64][33];
  __shared__ float Bs[32][65];
  const int tid  = threadIdx.x;
  const int wave = tid >> 5;
  const int lane = tid & 31;
  const int hi   = lane >> 4;   // 0 or 1 (lane half)
  const int lrow = lane & 15;
  const int mBlk = blockIdx.x * 64;
  const int nBlk = blockIdx.y * 64;

  v8f acc[4];
#pragma unroll
  for (int t = 0; t < 4; ++t) acc[t] = (v8f){0.f, 0.f, 0.f, 0.f, 0.f, 0.f, 0.f, 0.f};

  for (int kb = 0; kb < K; kb += 32) {
    // stage A tile 64x32 (coalesced: 32-float runs per row)
    for (int i = tid; i < 64 * 32; i += 128) {
      int r = i >> 5, c = i & 31;
      int gm = mBlk + r;
      As[r][c] = (gm < M) ? A[(long long)gm * lda + kb + c] : 0.f;
    }
    // stage B tile 32x64
    for (int i = tid; i < 32 * 64; i += 128) {
      int r = i >> 6, c = i & 63;
      int gn = nBlk + c;
      Bs[r][c] = (gn < Ncols) ? B[(long long)(kb + r) * ldb + gn] : 0.f;
    }
    __syncthreads();

    const int am = wave * 16 + lrow;
#pragma unroll
    for (int kk = 0; kk < 32; kk += 4) {
#if defined(__gfx1250__) && __has_builtin(__builtin_amdgcn_wmma_f32_16x16x4_f32)
      // A frag (16x4): lanes 0-15 hold K={kk,kk+1}, lanes 16-31 K={kk+2,kk+3}
      v2f a;
      a.x = As[am][kk + 2 * hi + 0];
      a.y = As[am][kk + 2 * hi + 1];
#pragma unroll
      for (int t = 0; t < 4; ++t) {
        v2f b;  // B frag (4x16), mirrored K striping across lane halves
        b.x = Bs[kk + 2 * hi + 0][t * 16 + lrow];
        b.y = Bs[kk + 2 * hi + 1][t * 16 + lrow];
        acc[t] = __builtin_amdgcn_wmma_f32_16x16x4_f32(
            false, a, false, b, (short)0, acc[t], false, false);
      }
#else
      // scalar fallback with identical math (keeps compile green if the
      // builtin signature differs; histogram will show wmma=0)
#pragma unroll
      for (int t = 0; t < 4; ++t) {
#pragma unroll
        for (int r = 0; r < 8; ++r) {
          int m = wave * 16 + hi * 8 + r;
          float s = acc[t][r];
#pragma unroll
          for (int k2 = 0; k2 < 4; ++k2)
            s += As[m][kk + k2] * Bs[kk + k2][t * 16 + lrow];
          acc[t][r] = s;
        }
      }
#endif
    }
    __syncthreads();
  }

  // C layout: VGPR r holds M = base + r (lanes 0-15) / base + r + 8 (16-31)
  const int cm  = mBlk + wave * 16 + hi * 8;
  const int cn0 = nBlk + lrow;
#pragma unroll
  for (int t = 0; t < 4; ++t) {
    int gc = cn0 + t * 16;
    if (gc < Ncols) {
#pragma unroll
      for (int r = 0; r < 8; ++r) {
        int gm = cm + r;
        if (gm < M) C[(long long)gm * ldc + gc] = acc[t][r];
      }
    }
  }
}

// ----------------- edge update: relu(bn(ea@We + Hd[dst] + Hs[src])) --------
__global__ void edge_update_k(const float* __restrict__ ea_in, int ea_stride, int in_e,
                              const float* __restrict__ euw, const float* __restrict__ eub,
                              const float* __restrict__ bg, const float* __restrict__ bb,
                              const float* __restrict__ bm, const float* __restrict__ bv,
                              const float* __restrict__ HdHs,
                              const int* __restrict__ src, const int* __restrict__ dst,
                              float* __restrict__ ea_out) {
  int idx = blockIdx.x * 128 + threadIdx.x;
  int e = idx >> 5, c = idx & 31;
  if (e >= N_EDGES) return;
  int s = src[e], d = dst[e];
  float t = eub[c];
  for (int j = 0; j < in_e; ++j)
    t += ea_in[(long long)e * ea_stride + j] * euw[j * 32 + c];
  t += HdHs[(long long)d * 64 + c] + HdHs[(long long)s * 64 + 32 + c];
  t = (t - bm[c]) * (bg[c] / sqrtf(bv[c] + BN_EPS)) + bb[c];
  ea_out[(long long)e * 32 + c] = fmaxf(t, 0.f);
}

// --------------------- seg-mean of ea over dst (loop_ea) -------------------
__global__ void segmean_acc_k(const float* __restrict__ ea, const int* __restrict__ dst,
                              float* __restrict__ loop_ea, float* __restrict__ cnt) {
  int idx = blockIdx.x * 128 + threadIdx.x;
  int e = idx >> 5, c = idx & 31;
  if (e >= N_EDGES) return;
  int d = dst[e];
  atomicAdd(&loop_ea[(long long)d * 32 + c], ea[(long long)e * 32 + c]);
  if (c == 0) atomicAdd(&cnt[d], 1.f);
}
__global__ void segmean_div_k(float* __restrict__ loop_ea, const float* __restrict__ cnt) {
  int idx = blockIdx.x * 256 + threadIdx.x;
  if (idx >= N_NODES * 32) return;
  loop_ea[idx] /= fmaxf(cnt[idx >> 5], 1.f);
}

// -------- fold gwe·aedge -> we_att[32][4];  (et never materialized) --------
__global__ void attn_fold_k(const float* __restrict__ gwe, const float* __restrict__ aedge,
                            float* __restrict__ watt) {
  int t = threadIdx.x;          // 128 threads: j = t/4, h = t%4
  int j = t >> 2, h = t & 3;
  float s = 0.f;
  for (int c = 0; c < 128; ++c)
    s += gwe[(long long)j * HDIM + h * 128 + c] * aedge[h * 128 + c];
  watt[j * 4 + h] = s;
}

// -------- per-node attention scalars a_src, a_dst from ht ------------------
__global__ void node_attn_k(const float* __restrict__ ht, const float* __restrict__ asrc,
                            const float* __restrict__ adst,
                            float* __restrict__ asn, float* __restrict__ adn) {
  int idx = blockIdx.x * 256 + threadIdx.x;
  if (idx >= N_NODES * 4) return;
  int n = idx >> 2, h = idx & 3;
  const float* row = ht + (long long)n * HDIM + h * 128;
  float sa = 0.f, sd = 0.f;
  for (int c = 0; c < 128; ++c) {
    float v = row[c];
    sa += v * asrc[h * 128 + c];
    sd += v * adst[h * 128 + c];
  }
  asn[idx] = sa;
  adn[idx] = sd;
}

// -------- logits (leaky 0.2) + segment max (real edges / self loops) -------
__global__ void logits_e_k(const float* __restrict__ ea, const float* __restrict__ watt,
                           const float* __restrict__ asn, const float* __restrict__ adn,
                           const int* __restrict__ src, const int* __restrict__ dst,
                           float* __restrict__ lg_e, float* __restrict__ segmax) {
  int idx = blockIdx.x * 256 + threadIdx.x;
  if (idx >= N_EDGES * 4) return;
  int e = idx >> 2, h = idx & 3;
  float ae = 0.f;
  for (int j = 0; j < 32; ++j) ae += ea[(long long)e * 32 + j] * watt[j * 4 + h];
  float lg = asn[src[e] * 4 + h] + adn[dst[e] * 4 + h] + ae;
  lg = (lg > 0.f) ? lg : 0.2f * lg;
  lg_e[idx] = lg;
  atomicMaxF(&segmax[dst[e] * 4 + h], lg);
}
__global__ void logits_n_k(const float* __restrict__ loop_ea, const float* __restrict__ watt,
                           const float* __restrict__ asn, const float* __restrict__ adn,
                           float* __restrict__ lg_n, float* __restrict__ segmax) {
  int idx = blockIdx.x * 256 + threadIdx.x;
  if (idx >= N_NODES * 4) return;
  int n = idx >> 2, h = idx & 3;
  float ae = 0.f;
  for (int j = 0; j < 32; ++j) ae += loop_ea[(long long)n * 32 + j] * watt[j * 4 + h];
  float lg = asn[idx] + adn[idx] + ae;
  lg = (lg > 0.f) ? lg : 0.2f * lg;
  lg_n[idx] = lg;
  atomicMaxF(&segmax[idx], lg);
}

// -------- exp(logit - max) + segment sum (in-place over logit bufs) --------
__global__ void expsum_e_k(float* __restrict__ lg_e, const int* __restrict__ dst,
                           const float* __restrict__ segmax, float* __restrict__ segsum) {
  int idx = blockIdx.x * 256 + threadIdx.x;
  if (idx >= N_EDGES * 4) return;
  int e = idx >> 2, h = idx & 3;
  float ev = expf(lg_e[idx] - segmax[dst[e] * 4 + h]);
  lg_e[idx] = ev;
  atomicAdd(&segsum[dst[e] * 4 + h], ev);
}
__global__ void expsum_n_k(float* __restrict__ lg_n, const float* __restrict__ segmax,
                           float* __restrict__ segsum) {
  int idx = blockIdx.x * 256 + threadIdx.x;
  if (idx >= N_NODES * 4) return;
  float ev = expf(lg_n[idx] - segmax[idx]);
  lg_n[idx] = ev;
  atomicAdd(&segsum[idx], ev);
}

// -------- message aggregation: agg[dst] += alpha * ht[src] -----------------
__global__ void agg_e_k(const float* __restrict__ ht, const float* __restrict__ lg_e,
                        const float* __restrict__ segsum,
                        const int* __restrict__ src, const int* __restrict__ dst,
                        float* __restrict__ agg) {
  long long idx = (long long)blockIdx.x * 256 + threadIdx.x;
  if (idx >= (long long)N_EDGES * HDIM) return;
  int e = (int)(idx >> 9), c = (int)(idx & 511), h = c >> 7;
  int d = dst[e], s = src[e];
  float alpha = lg_e[e * 4 + h] / segsum[d * 4 + h];
  atomicAdd(&agg[(long long)d * HDIM + c], alpha * ht[(long long)s * HDIM + c]);
}
// self-loops run after agg_e on the stream; each (n,c) unique -> plain add
__global__ void agg_n_k(const float* __restrict__ ht, const float* __restrict__ lg_n,
                        const float* __restrict__ segsum, float* __restrict__ agg) {
  long long idx = (long long)blockIdx.x * 256 + threadIdx.x;
  if (idx >= (long long)N_NODES * HDIM) return;
  int n = (int)(idx >> 9), c = (int)(idx & 511), h = c >> 7;
  float alpha = lg_n[n * 4 + h] / segsum[n * 4 + h];
  agg[idx] += alpha * ht[idx];
}

// -------- node BN + ReLU: h = relu(bn(agg + gb)) ---------------------------
__global__ void bn_relu_node_k(const float* __restrict__ agg, const float* __restrict__ gb,
                               const float* __restrict__ bg, const float* __restrict__ bb,
                               const float* __restrict__ bm, const float* __restrict__ bv,
                               float* __restrict__ hout) {
  long long idx = (long long)blockIdx.x * 256 + threadIdx.x;
  if (idx >= (long long)N_NODES * HDIM) return;
  int c = (int)(idx & 511);
  float t = agg[idx] + gb[c];
  t = (t - bm[c]) * (bg[c] / sqrtf(bv[c] + BN_EPS)) + bb[c];
  hout[idx] = fmaxf(t, 0.f);
}

// -------- graph mean-pool --------------------------------------------------
__global__ void pool_acc_k(const float* __restrict__ h, const int* __restrict__ batch,
                           float* __restrict__ pool, float* __restrict__ cntg) {
  long long idx = (long long)blockIdx.x * 256 + threadIdx.x;
  if (idx >= (long long)N_NODES * HDIM) return;
  int n = (int)(idx >> 9), c = (int)(idx & 511);
  int g = batch[n];
  atomicAdd(&pool[(long long)g * HDIM + c], h[idx]);
  if (c == 0) atomicAdd(&cntg[g], 1.f);
}
__global__ void pool_div_k(float* __restrict__ pool, const float* __restrict__ cntg) {
  int idx = blockIdx.x * 256 + threadIdx.x;
  if (idx >= NGRAPH * HDIM) return;
  pool[idx] /= fmaxf(cntg[idx >> 9], 1.f);
}

__global__ void bias_relu_k(float* __restrict__ hid, const float* __restrict__ b) {
  int idx = blockIdx.x * 256 + threadIdx.x;
  if (idx >= NGRAPH * HDIM) return;
  hid[idx] = fmaxf(hid[idx] + b[idx & 511], 0.f);
}

__global__ void out_k(const float* __restrict__ hid, const float* __restrict__ ow,
                      const float* __restrict__ ob, float* __restrict__ out) {
  int g = threadIdx.x;
  if (g >= NGRAPH) return;
  float s = ob[0];
  for (int c = 0; c < HDIM; ++c) s += hid[(long long)g * HDIM + c] * ow[c];
  out[g] = s;
}

// ===========================================================================
struct LayerP {
  const float *euw, *eub, *eubn_g, *eubn_b, *eubn_m, *eubn_v;
  const float *gw, *gb, *asrc, *adst, *aedge, *gwe;
  const float *bn_g, *bn_b, *bn_m, *bn_v;
};

static inline int cdiv(long long a, long long b) { return (int)((a + b - 1) / b); }

extern "C" void kernel_launch(void* const* d_in, const int* in_sizes, int n_in,
                              void* d_out, int out_size, void* d_ws, size_t ws_size,
                              hipStream_t stream) {
  const float* x         = (const float*)d_in[0];
  const int*   x_emb     = (const int*)d_in[1];
  const int*   eidx      = (const int*)d_in[2];
  const float* edge_attr = (const float*)d_in[3];
  const int*   batch     = (const int*)d_in[4];
  const int*   src = eidx;
  const int*   dst = eidx + N_EDGES;

  // ---- param leaf layout: detect jax sorted-pytree vs insertion order ----
  const float *emb, *dense_w, *dense_b, *out_w, *out_b;
  LayerP L[3];
  bool sorted = (n_in > 5 && in_sizes[5] == 512);  // sorted: dense_b first
  if (sorted) {
    dense_b = (const float*)d_in[5];
    dense_w = (const float*)d_in[6];
    emb     = (const float*)d_in[7];
    int p = 8;
    for (int l = 0; l < 3; ++l) {
      L[l].adst   = (const float*)d_in[p++];
      L[l].aedge  = (const float*)d_in[p++];
      L[l].asrc   = (const float*)d_in[p++];
      L[l].bn_b   = (const float*)d_in[p++];
      L[l].bn_g   = (const float*)d_in[p++];
      L[l].bn_m   = (const float*)d_in[p++];
      L[l].bn_v   = (const float*)d_in[p++];
      L[l].eub    = (const float*)d_in[p++];
      L[l].eubn_b = (const float*)d_in[p++];
      L[l].eubn_g = (const float*)d_in[p++];
      L[l].eubn_m = (const float*)d_in[p++];
      L[l].eubn_v = (const float*)d_in[p++];
      L[l].euw    = (const float*)d_in[p++];
      L[l].gb     = (const float*)d_in[p++];
      L[l].gw     = (const float*)d_in[p++];
      L[l].gwe    = (const float*)d_in[p++];
    }
    out_b = (const float*)d_in[56];
    out_w = (const float*)d_in[57];
  } else {
    emb = (const float*)d_in[5];
    int p = 6;
    for (int l = 0; l < 3; ++l) {
      L[l].euw    = (const float*)d_in[p++];
      L[l].eub    = (const float*)d_in[p++];
      L[l].eubn_g = (const float*)d_in[p++];
      L[l].eubn_b = (const float*)d_in[p++];
      L[l].eubn_m = (const float*)d_in[p++];
      L[l].eubn_v = (const float*)d_in[p++];
      L[l].gw     = (const float*)d_in[p++];
      L[l].gb     = (const float*)d_in[p++];
      L[l].asrc   = (const float*)d_in[p++];
      L[l].adst   = (const float*)d_in[p++];
      L[l].aedge  = (const float*)d_in[p++];
      L[l].gwe    = (const float*)d_in[p++];
      L[l].bn_g   = (const float*)d_in[p++];
      L[l].bn_b   = (const float*)d_in[p++];
      L[l].bn_m   = (const float*)d_in[p++];
      L[l].bn_v   = (const float*)d_in[p++];
    }
    dense_w = (const float*)d_in[54];
    dense_b = (const float*)d_in[55];
    out_w   = (const float*)d_in[56];
    out_b   = (const float*)d_in[57];
  }

  // ---- workspace carve-out (floats) ----
  float* ws = (float*)d_ws;
  size_t o = 0;
  float* hA      = ws + o; o += (size_t)N_NODES * 640;   // layer0 h (lda 640), reused
  float* hB      = ws + o; o += (size_t)N_NODES * 640;
  float* eaA     = ws + o; o += (size_t)N_EDGES * 32;
  float* eaB     = ws + o; o += (size_t)N_EDGES * 32;
  float* gwpad   = ws + o; o += (size_t)640 * HDIM;
  float* w2pad   = ws + o; o += (size_t)640 * 64;
  float* HdHs    = ws + o; o += (size_t)N_NODES * 64;
  float* ht      = ws + o; o += (size_t)N_NODES * HDIM;
  float* loop_ea = ws + o; o += (size_t)N_NODES * 32;
  float* cnt     = ws + o; o += (size_t)N_NODES;
  float* asn     = ws + o; o += (size_t)N_NODES * 4;
  float* adn     = ws + o; o += (size_t)N_NODES * 4;
  float* watt    = ws + o; o += 128;
  float* lg_e    = ws + o; o += (size_t)N_EDGES * 4;
  float* lg_n    = ws + o; o += (size_t)N_NODES * 4;
  float* segmax  = ws + o; o += (size_t)N_NODES * 4;
  float* segsum  = ws + o; o += (size_t)N_NODES * 4;
  float* agg     = ws + o; o += (size_t)N_NODES * HDIM;
  float* pool    = ws + o; o += (size_t)NGRAPH * HDIM;
  float* cntg    = ws + o; o += (size_t)NGRAPH;
  float* hid     = ws + o; o += (size_t)NGRAPH * HDIM;
  (void)ws_size; (void)out_size;

  // ---- build h0 = [emb | x | 0] ----
  build_h0_k<<<cdiv((long long)N_NODES * 640, 256), 256, 0, stream>>>(x, x_emb, emb, hA);

  const int in_n_arr[3] = {622, 512, 512};
  const int in_e_arr[3] = {1, 32, 32};
  const int kpad_arr[3] = {640, 512, 512};
  float* hin_arr[3]  = {hA, hB, hA};
  const int hlda[3]  = {640, 512, 512};
  float* hout_arr[3] = {hB, hA, hB};
  const float* eain_arr[3] = {edge_attr, eaA, eaB};
  const int east_arr[3]    = {1, 32, 32};
  float* eaout_arr[3]      = {eaA, eaB, eaA};

  for (int l = 0; l < 3; ++l) {
    const LayerP& P = L[l];
    const int in_n = in_n_arr[l], in_e = in_e_arr[l], kpad = kpad_arr[l];
    float* hin  = hin_arr[l];
    float* hout = hout_arr[l];

    // pack weights (zero-padded K)
    pack_w2_k<<<cdiv((long long)kpad * 64, 256), 256, 0, stream>>>(P.euw, in_e, in_n, kpad, w2pad);
    pack_gw_k<<<cdiv((long long)kpad * HDIM, 256), 256, 0, stream>>>(P.gw, in_n, kpad, gwpad);

    // HdHs = h @ [Wd|Ws]   (N x 64)
    {
      dim3 g(cdiv(N_NODES, 64), 1);
      gemm_f32_wmma<<<g, 128, 0, stream>>>(hin, hlda[l], w2pad, 64, HdHs, 64,
                                           N_NODES, 64, kpad);
    }
    // ht = h @ gw          (N x 512)
    {
      dim3 g(cdiv(N_NODES, 64), cdiv(HDIM, 64));
      gemm_f32_wmma<<<g, 128, 0, stream>>>(hin, hlda[l], gwpad, HDIM, ht, HDIM,
                                           N_NODES, HDIM, kpad);
    }

    // edge update -> ea_out
    edge_update_k<<<cdiv((long long)N_EDGES * 32, 128), 128, 0, stream>>>(
        eain_arr[l], east_arr[l], in_e, P.euw, P.eub,
        P.eubn_g, P.eubn_b, P.eubn_m, P.eubn_v, HdHs, src, dst, eaout_arr[l]);
    const float* ea = eaout_arr[l];

    // loop_ea = seg_mean(ea, dst)
    fill_k<<<cdiv((long long)N_NODES * 32, 256), 256, 0, stream>>>(loop_ea, 0.f, (long long)N_NODES * 32);
    fill_k<<<cdiv(N_NODES, 256), 256, 0, stream>>>(cnt, 0.f, N_NODES);
    segmean_acc_k<<<cdiv((long long)N_EDGES * 32, 128), 128, 0, stream>>>(ea, dst, loop_ea, cnt);
    segmean_div_k<<<cdiv((long long)N_NODES * 32, 256), 256, 0, stream>>>(loop_ea, cnt);

    // attention scalars
    attn_fold_k<<<1, 128, 0, stream>>>(P.gwe, P.aedge, watt);
    node_attn_k<<<cdiv((long long)N_NODES * 4, 256), 256, 0, stream>>>(ht, P.asrc, P.adst, asn, adn);

    // segment softmax over dst (edges + self loops)
    fill_k<<<cdiv((long long)N_NODES * 4, 256), 256, 0, stream>>>(segmax, -3.0e38f, (long long)N_NODES * 4);
    fill_k<<<cdiv((long long)N_NODES * 4, 256), 256, 0, stream>>>(segsum, 0.f, (long long)N_NODES * 4);
    logits_e_k<<<cdiv((long long)N_EDGES * 4, 256), 256, 0, stream>>>(ea, watt, asn, adn, src, dst, lg_e, segmax);
    logits_n_k<<<cdiv((long long)N_NODES * 4, 256), 256, 0, stream>>>(loop_ea, watt, asn, adn, lg_n, segmax);
    expsum_e_k<<<cdiv((long long)N_EDGES * 4, 256), 256, 0, stream>>>(lg_e, dst, segmax, segsum);
    expsum_n_k<<<cdiv((long long)N_NODES * 4, 256), 256, 0, stream>>>(lg_n, segmax, segsum);

    // aggregate alpha * ht[src] into agg[dst]
    fill_k<<<cdiv((long long)N_NODES * HDIM, 256), 256, 0, stream>>>(agg, 0.f, (long long)N_NODES * HDIM);
    agg_e_k<<<cdiv((long long)N_EDGES * HDIM, 256), 256, 0, stream>>>(ht, lg_e, segsum, src, dst, agg);
    agg_n_k<<<cdiv((long long)N_NODES * HDIM, 256), 256, 0, stream>>>(ht, lg_n, segsum, agg);

    // h_next = relu(bn(agg + gb))
    bn_relu_node_k<<<cdiv((long long)N_NODES * HDIM, 256), 256, 0, stream>>>(
        agg, P.gb, P.bn_g, P.bn_b, P.bn_m, P.bn_v, hout);
  }

  // ---- global mean pool + dense + out ----
  float* hfin = hout_arr[2];  // hB, stride 512
  fill_k<<<cdiv((long long)NGRAPH * HDIM, 256), 256, 0, stream>>>(pool, 0.f, (long long)NGRAPH * HDIM);
  fill_k<<<1, 256, 0, stream>>>(cntg, 0.f, NGRAPH);
  pool_acc_k<<<cdiv((long long)N_NODES * HDIM, 256), 256, 0, stream>>>(hfin, batch, pool, cntg);
  pool_div_k<<<cdiv((long long)NGRAPH * HDIM, 256), 256, 0, stream>>>(pool, cntg);

  {
    dim3 g(cdiv(NGRAPH, 64), cdiv(HDIM, 64));
    gemm_f32_wmma<<<g, 128, 0, stream>>>(pool, HDIM, dense_w, HDIM, hid, HDIM,
                                         NGRAPH, HDIM, HDIM);
  }
  bias_relu_k<<<cdiv((long long)NGRAPH * HDIM, 256), 256, 0, stream>>>(hid, dense_b);
  out_k<<<1, 64, 0, stream>>>(hid, out_w, out_b, (float*)d_out);
}